// AtomicMoment_47493748359214
// MI455X (gfx1250) — compile-verified
//
#include <hip/hip_runtime.h>
#include <hip/hip_bf16.h>

#define N_U      8
#define N_ATOMS  20000
#define N_EDGES  640000
#define N_CHEB   9
#define N_PATHS  11
#define WAVES_PER_BLOCK 4
#define EDGE_BLOCK (WAVES_PER_BLOCK * 32)

// accumulator layout inside d_ws (floats):
//   acc0: [u][atom]            offset 0        size 160000
//   acc1: [u][atom][a]         offset 160000   size 480000
//   acc2: [u][atom][a*3+b]     offset 640000   size 1440000
#define ACC1_OFF 160000
#define ACC2_OFF 640000
#define ACC_TOTAL 2080000

typedef __attribute__((ext_vector_type(2))) float v2f;
typedef __attribute__((ext_vector_type(8))) float v8f;

// silu via native v_exp_f32 + v_rcp_f32 (no IEEE div_scale/div_fixup chain)
__device__ __forceinline__ float silu_fast(float x) {
    return x * __builtin_amdgcn_rcpf(1.0f + __expf(-x));
}

// One 16x8 @ 8x8 matmul tile via two V_WMMA_F32_16X16X4_F32 (K = 4+4).
// src: 16 rows x 8 cols in LDS (one 16-edge tile). W: row-major 8x8.
// B columns 8..15 are replicas of 0..7 (unconditional loads, no EXEC
// masking); the duplicate D columns are simply never read back.
__device__ __forceinline__ v8f mm8_tile(const float (*__restrict__ src)[N_U],
                                        const float* __restrict__ W, int lane) {
    const int half = lane >> 4;       // selects K pair {k,k+1} vs {k+2,k+3}
    const int mrow = lane & 15;       // A row (edge within tile)
    const int nc   = lane & 7;        // B/D column, replicated mod 8
    v8f c = {};
#pragma unroll
    for (int kb = 0; kb < N_U; kb += 4) {
        const int k0 = kb + 2 * half;
        v2f a, b;
        a.x = src[mrow][k0];
        a.y = src[mrow][k0 + 1];
        b.x = W[(k0    ) * N_U + nc];
        b.y = W[(k0 + 1) * N_U + nc];
        c = __builtin_amdgcn_wmma_f32_16x16x4_f32(false, a, false, b,
                                                  (short)0, c, false, false);
    }
    return c;
}

// Store D (16x16, cols 0..7 valid) as activations back to LDS: dst[row][col].
// Bias + activation computed unconditionally (SIMD lockstep — masking saves
// nothing); only the LDS stores are guarded (lanes n and n+8 would race).
__device__ __forceinline__ void store_act(float (*__restrict__ dst)[N_U], v8f c,
                                          const float* __restrict__ B, int lane,
                                          bool act) {
    const int half = lane >> 4;
    const int ncol = lane & 15;
    const float bias = B[lane & 7];
    float v[8];
#pragma unroll
    for (int g = 0; g < 8; ++g) {
        float t = c[g] + bias;
        v[g] = act ? silu_fast(t) : t;
    }
    if (ncol < N_U) {
#pragma unroll
        for (int g = 0; g < 8; ++g) dst[g + 8 * half][ncol] = v[g];
    }
}

__global__ void zero_acc_kernel(float* __restrict__ acc) {
    int i = blockIdx.x * blockDim.x + threadIdx.x;
    if (i < ACC_TOTAL) acc[i] = 0.0f;
}

__global__ __launch_bounds__(EDGE_BLOCK)
void edge_kernel(const float* __restrict__ edge_vector,
                 const int*   __restrict__ edge_idx,
                 const int*   __restrict__ atom_type,
                 const float* __restrict__ feats0,
                 const float* __restrict__ feats1,
                 const float* __restrict__ feats2,
                 const float* __restrict__ W_rad,
                 const float* __restrict__ mlp_w1, const float* __restrict__ mlp_b1,
                 const float* __restrict__ mlp_w2, const float* __restrict__ mlp_b2,
                 const float* __restrict__ mlp_w3, const float* __restrict__ mlp_b3,
                 const float* __restrict__ lp0,
                 const float* __restrict__ lp1,
                 const float* __restrict__ lp2,
                 float* __restrict__ acc) {
    __shared__ float fu_s[WAVES_PER_BLOCK][32][N_U];
    __shared__ float hA[WAVES_PER_BLOCK][32][N_U];
    __shared__ float hB[WAVES_PER_BLOCK][32][N_U];
    __shared__ float Rw_s[WAVES_PER_BLOCK][N_PATHS][32][N_U];

    const int lane = threadIdx.x & 31;
    const int w    = threadIdx.x >> 5;
    const int e    = blockIdx.x * EDGE_BLOCK + w * 32 + lane;

    // ---------------- Phase 1: per-edge radial basis -> fu (LDS) -------------
    const float evx = edge_vector[3 * e + 0];
    const float evy = edge_vector[3 * e + 1];
    const float evz = edge_vector[3 * e + 2];
    const float r   = sqrtf(evx * evx + evy * evy + evz * evz);
    const float d   = r * (1.0f / 5.0f);             // r / R_CUT
    float x = 2.0f * d - 1.0f;
    x = fminf(1.0f, fmaxf(-1.0f, x));

    float T[N_CHEB];
    T[0] = 1.0f; T[1] = x;
#pragma unroll
    for (int k = 2; k < N_CHEB; ++k) T[k] = 2.0f * x * T[k - 1] - T[k - 2];

    const float d6 = d * d * d * d * d * d;
    float env = 1.0f - 28.0f * d6 + 48.0f * d6 * d - 21.0f * d6 * d * d;
    env = (d < 1.0f) ? env : 0.0f;

    const int ia = edge_idx[e];
    const int ja = edge_idx[N_EDGES + e];
    const int ti = atom_type[ia];
    const int tj = atom_type[ja];
    const float* __restrict__ Wij = W_rad + (ti * 8 + tj) * (N_CHEB * N_U);

#pragma unroll
    for (int u = 0; u < N_U; ++u) {
        float s = 0.0f;
#pragma unroll
        for (int k = 0; k < N_CHEB; ++k) s += T[k] * Wij[k * N_U + u];
        fu_s[w][lane][u] = env * s;
    }
    const float rinv = 1.0f / r;
    const float ux = evx * rinv, uy = evy * rinv, uz = evz * rinv;
    __syncthreads();

    // ---------------- Phase 2: 11 path-MLPs via WMMA -------------------------
    for (int p = 0; p < N_PATHS; ++p) {
        const float* W1 = mlp_w1 + p * 64; const float* B1 = mlp_b1 + p * 8;
        const float* W2 = mlp_w2 + p * 64; const float* B2 = mlp_b2 + p * 8;
        const float* W3 = mlp_w3 + p * 64; const float* B3 = mlp_b3 + p * 8;

        // layer 1: fu -> hA
        {
            v8f c0 = mm8_tile(&fu_s[w][0],  W1, lane);
            v8f c1 = mm8_tile(&fu_s[w][16], W1, lane);
            store_act(&hA[w][0],  c0, B1, lane, true);
            store_act(&hA[w][16], c1, B1, lane, true);
        }
        __syncthreads();
        // layer 2: hA -> hB
        {
            v8f c0 = mm8_tile(&hA[w][0],  W2, lane);
            v8f c1 = mm8_tile(&hA[w][16], W2, lane);
            store_act(&hB[w][0],  c0, B2, lane, true);
            store_act(&hB[w][16], c1, B2, lane, true);
        }
        __syncthreads();
        // layer 3: hB -> Rw (fold in lp path weight, no activation)
        {
            v8f c0 = mm8_tile(&hB[w][0],  W3, lane);
            v8f c1 = mm8_tile(&hB[w][16], W3, lane);
            const int half = lane >> 4;
            const int ncol = lane & 15;
            const int nc   = lane & 7;
            const float* lpp; int lr;
            if (p < 3)      { lpp = lp0; lr = p; }
            else if (p < 7) { lpp = lp1; lr = p - 3; }
            else            { lpp = lp2; lr = p - 7; }
            const float lpv  = lpp[lr * N_U + nc];   // unconditional
            const float bias = B3[nc];               // unconditional
            float v0[8], v1[8];
#pragma unroll
            for (int g = 0; g < 8; ++g) {
                v0[g] = (c0[g] + bias) * lpv;
                v1[g] = (c1[g] + bias) * lpv;
            }
            if (ncol < N_U) {
#pragma unroll
                for (int g = 0; g < 8; ++g) {
                    Rw_s[w][p][ 0 + g + 8 * half][ncol] = v0[g];
                    Rw_s[w][p][16 + g + 8 * half][ncol] = v1[g];
                }
            }
        }
        __syncthreads();
    }

    // ---------------- Phase 3: tensor products + atomic scatter --------------
    float* __restrict__ acc0 = acc;
    float* __restrict__ acc1 = acc + ACC1_OFF;
    float* __restrict__ acc2 = acc + ACC2_OFF;
    const float un[3] = {ux, uy, uz};

#pragma unroll 2
    for (int u = 0; u < N_U; ++u) {
        float Rw[N_PATHS];
#pragma unroll
        for (int p = 0; p < N_PATHS; ++p) Rw[p] = Rw_s[w][p][lane][u];

        const float f0 = feats0[u * N_ATOMS + ja];
        float f1v[3], f2v[9];
#pragma unroll
        for (int a = 0; a < 3; ++a) f1v[a] = feats1[(u * N_ATOMS + ja) * 3 + a];
#pragma unroll
        for (int k = 0; k < 9; ++k) f2v[k] = feats2[(u * N_ATOMS + ja) * 9 + k];

        const float s1 = f1v[0] * ux + f1v[1] * uy + f1v[2] * uz;   // f1 . unit
        float g_[3];
#pragma unroll
        for (int a = 0; a < 3; ++a)                                  // f2 @ unit
            g_[a] = f2v[a * 3 + 0] * ux + f2v[a * 3 + 1] * uy + f2v[a * 3 + 2] * uz;
        const float q = g_[0] * ux + g_[1] * uy + g_[2] * uz;        // u^T f2 u

        // rank 0 message (paths 0,1,2)
        atomicAdd(&acc0[u * N_ATOMS + ia],
                  Rw[0] * f0 + Rw[1] * s1 + Rw[2] * q);

        // rank 1 message (paths 3,4,5,6)
#pragma unroll
        for (int a = 0; a < 3; ++a) {
            const float m1 = Rw[3] * f0 * un[a] + Rw[4] * f1v[a] +
                             Rw[5] * s1 * un[a] + Rw[6] * g_[a];
            atomicAdd(&acc1[(u * N_ATOMS + ia) * 3 + a], m1);
        }

        // rank 2 message (paths 7,8,9,10)
#pragma unroll
        for (int a = 0; a < 3; ++a) {
            const float r9  = Rw[9]  * f1v[a];
            const float r7  = Rw[7]  * f0 * un[a];
            const float r10 = Rw[10] * g_[a];
#pragma unroll
            for (int b = 0; b < 3; ++b) {
                const float m2 = r7 * un[b] + Rw[8] * f2v[a * 3 + b] +
                                 r9 * un[b] + r10 * un[b];
                atomicAdd(&acc2[(u * N_ATOMS + ia) * 9 + a * 3 + b], m2);
            }
        }
    }
}

// out[c,...] = inv * sum_u lc[c,u] * acc[u,...]
__global__ void mix_kernel(const float* __restrict__ acc,
                           const float* __restrict__ lc0,
                           const float* __restrict__ lc1,
                           const float* __restrict__ lc2,
                           float* __restrict__ out) {
    const int tid = blockIdx.x * blockDim.x + threadIdx.x;
    if (tid >= N_ATOMS * 13) return;
    const int atom = tid / 13;
    const int comp = tid % 13;
    const float inv = 0.17677669529663687f;   // 1/sqrt(32)

    const float* lc;
    long in_base, out_base;
    long stride_unit;   // per-u / per-c stride in elements for this region
    if (comp == 0) {
        lc = lc0;
        in_base  = (long)atom;
        out_base = (long)atom;
        stride_unit = (long)N_ATOMS;
    } else if (comp <= 3) {
        lc = lc1;
        const int a = comp - 1;
        in_base  = ACC1_OFF + (long)atom * 3 + a;
        out_base = ACC1_OFF + (long)atom * 3 + a;
        stride_unit = (long)N_ATOMS * 3;
    } else {
        lc = lc2;
        const int k = comp - 4;
        in_base  = ACC2_OFF + (long)atom * 9 + k;
        out_base = ACC2_OFF + (long)atom * 9 + k;
        stride_unit = (long)N_ATOMS * 9;
    }

    float vals[N_U];
#pragma unroll
    for (int u = 0; u < N_U; ++u) vals[u] = acc[in_base + u * stride_unit];

#pragma unroll
    for (int c = 0; c < N_U; ++c) {
        float s = 0.0f;
#pragma unroll
        for (int u = 0; u < N_U; ++u) s += lc[c * N_U + u] * vals[u];
        out[out_base + c * stride_unit] = inv * s;
    }
}

extern "C" void kernel_launch(void* const* d_in, const int* in_sizes, int n_in,
                              void* d_out, int out_size, void* d_ws, size_t ws_size,
                              hipStream_t stream) {
    const float* edge_vector = (const float*)d_in[0];
    const int*   edge_idx    = (const int*)  d_in[1];
    const int*   atom_type   = (const int*)  d_in[2];
    const float* feats0      = (const float*)d_in[3];
    const float* feats1      = (const float*)d_in[4];
    const float* feats2      = (const float*)d_in[5];
    const float* W_rad       = (const float*)d_in[6];
    const float* mlp_w1      = (const float*)d_in[7];
    const float* mlp_b1      = (const float*)d_in[8];
    const float* mlp_w2      = (const float*)d_in[9];
    const float* mlp_b2      = (const float*)d_in[10];
    const float* mlp_w3      = (const float*)d_in[11];
    const float* mlp_b3      = (const float*)d_in[12];
    const float* lp0         = (const float*)d_in[13];
    const float* lp1         = (const float*)d_in[14];
    const float* lp2         = (const float*)d_in[15];
    const float* lc0         = (const float*)d_in[16];
    const float* lc1         = (const float*)d_in[17];
    const float* lc2         = (const float*)d_in[18];

    float* acc = (float*)d_ws;
    float* out = (float*)d_out;

    zero_acc_kernel<<<(ACC_TOTAL + 255) / 256, 256, 0, stream>>>(acc);

    edge_kernel<<<N_EDGES / EDGE_BLOCK, EDGE_BLOCK, 0, stream>>>(
        edge_vector, edge_idx, atom_type, feats0, feats1, feats2, W_rad,
        mlp_w1, mlp_b1, mlp_w2, mlp_b2, mlp_w3, mlp_b3,
        lp0, lp1, lp2, acc);

    mix_kernel<<<(N_ATOMS * 13 + 255) / 256, 256, 0, stream>>>(
        acc, lc0, lc1, lc2, out);
}